// WindowingLayer_13022340841695
// MI455X (gfx1250) — compile-verified
//
#include <hip/hip_runtime.h>
#include <stdint.h>

typedef float v4f __attribute__((ext_vector_type(4)));
typedef int   v4i __attribute__((ext_vector_type(4)));
typedef __attribute__((address_space(1))) v4i* gv4i_p;  // global int4*
typedef __attribute__((address_space(3))) v4i* lv4i_p;  // LDS int4*

#define WIN   16
#define FEA   64
#define SEQ   2048
#define BATCH 32
#define TS    64                    // output s-positions per block
#define NROWS (TS + WIN)            // 80 staged input rows (79 needed; 80 = even 256-thread split)
#define NT4   (NROWS * (FEA / 4))   // 1280 float4s in LDS = 20 KB

// out[b][s][w][f] = in[b][s+w][f], zero when s+w >= SEQ.
// Stage input rows [s0, s0+NROWS) into LDS once (async path, ASYNCcnt),
// then fan out TS * WIN * FEA floats as contiguous non-temporal 128-bit stores.
__global__ __launch_bounds__(256) void window_stage_kernel(
    const float* __restrict__ in, float* __restrict__ out) {
  __shared__ __align__(16) v4f smem[NT4];

  const int tid = threadIdx.x;
  const int s0  = blockIdx.x * TS;
  const int b   = blockIdx.y;

  const float* inb = in + ((size_t)b * SEQ + (size_t)s0) * FEA;

  // ---- stage phase: 1280 float4 -> LDS; zero-fill rows past end of sequence ----
#pragma unroll
  for (int k = 0; k < NT4 / 256; ++k) {
    const int idx = tid + k * 256;   // float4 index within staged tile
    const int row = idx >> 4;        // input row within tile (16 float4 per row)
    if (s0 + row < SEQ) {
#if __has_builtin(__builtin_amdgcn_global_load_async_to_lds_b128)
      __builtin_amdgcn_global_load_async_to_lds_b128(
          (gv4i_p)(inb + (size_t)idx * 4),
          (lv4i_p)&smem[idx],
          /*offset=*/0, /*cpol=*/0);
#else
      const unsigned lds_off = (unsigned)(uintptr_t)&smem[idx]; // low 32 bits = LDS byte addr
      const unsigned g_off   = (unsigned)idx * 16u;
      asm volatile("global_load_async_to_lds_b128 %0, %1, %2"
                   :: "v"(lds_off), "v"(g_off), "s"(inb)
                   : "memory");
#endif
    } else {
      smem[idx] = (v4f){0.f, 0.f, 0.f, 0.f};  // zero padding for tail windows
    }
  }
#if __has_builtin(__builtin_amdgcn_s_wait_asynccnt)
  __builtin_amdgcn_s_wait_asynccnt(0);
#else
  asm volatile("s_wait_asynccnt 0x0" ::: "memory");
#endif
  __syncthreads();

  // ---- emit phase: one full output s-position (4 KB, contiguous) per iteration ----
  v4f* outv = (v4f*)out + ((size_t)b * SEQ + (size_t)s0) * (WIN * FEA / 4);
  const int w  = tid >> 4;   // window offset 0..15
  const int f4 = tid & 15;   // float4 within feature row
#pragma unroll 4
  for (int sl = 0; sl < TS; ++sl) {
    v4f v = smem[(sl + w) * (FEA / 4) + f4];
    __builtin_nontemporal_store(v, &outv[(size_t)sl * (WIN * FEA / 4) + tid]);
  }
}

extern "C" void kernel_launch(void* const* d_in, const int* in_sizes, int n_in,
                              void* d_out, int out_size, void* d_ws, size_t ws_size,
                              hipStream_t stream) {
  (void)in_sizes; (void)n_in; (void)d_ws; (void)ws_size; (void)out_size;
  const float* in = (const float*)d_in[0];
  float* out = (float*)d_out;
  dim3 grid(SEQ / TS, BATCH);   // 32 x 32 blocks
  dim3 block(256);
  window_stage_kernel<<<grid, block, 0, stream>>>(in, out);
}